// SimpleTM_54443005444894
// MI455X (gfx1250) — compile-verified
//
#include <hip/hip_runtime.h>
#include <math.h>

typedef float v2f __attribute__((ext_vector_type(2)));
typedef float v8f __attribute__((ext_vector_type(8)));

#define NLOC 20000
#define NREP 64
#define MNB  30
#define KP   32          // K padded to 32 (cols 30,31 are zero)
#define XS_S 33          // LDS stride for xs (odd -> conflict-free fragment loads)
#define G_S  65          // LDS stride for g

__global__ __launch_bounds__(256)
void vecchia_loc_kernel(const float* __restrict__ locs,
                        const float* __restrict__ response,
                        const int*   __restrict__ cset,
                        const float* __restrict__ theta,
                        float* __restrict__ partial)
{
    __shared__ float xs[NREP * XS_S];   // scaled neighbor matrix, 64 x 32
    __shared__ float g [NREP * G_S];    // kernel matrix / Cholesky factor, 64 x 64
    __shared__ float ss[NREP];          // row sums of xs^2
    __shared__ float yv[NREP];          // response at this location (RHS)
    __shared__ float sc[KP];            // per-k scaling
    __shared__ float scal[4];           // nug_mean, sigma^2, 1/len^2, 1/nug_mean

    const int tid = threadIdx.x;
    const int loc = blockIdx.x;

    // ---- per-location scalars (thread 0) ----
    if (tid == 0) {
        float t0 = theta[0], t1 = theta[1];
        float t3 = theta[3], t4 = theta[4], t5 = theta[5];
        // scal0 = d0^2 / d4 ; d_j = dist(locs[j+1], locs[j])
        float dx0 = locs[2] - locs[0],  dy0 = locs[3] - locs[1];
        float d0  = sqrtf(dx0*dx0 + dy0*dy0);
        float dx4 = locs[10] - locs[8], dy4 = locs[11] - locs[9];
        float d4  = sqrtf(dx4*dx4 + dy4*dy4);
        float scal0 = d0*d0 / d4;
        float scale;
        if (loc == 0) scale = 1.0f;
        else {
            float ax = locs[2*loc]   - locs[2*(loc-1)];
            float ay = locs[2*loc+1] - locs[2*(loc-1)+1];
            scale = sqrtf(ax*ax + ay*ay) / scal0;
        }
        float ls  = logf(scale);
        float nug = expf(ls*t1 + t0);
        nug = fmaxf(nug - 1e-5f, 0.0f) + 1e-5f;          // relu(x-eps)+eps
        float sig = expf(ls*t4 + t3);
        float len = expf(t5) * sqrtf(3.0f);               // sqrt(2*SMOOTH)=sqrt(3)
        scal[0] = nug;
        scal[1] = sig * sig;
        scal[2] = 1.0f / (len * len);
        scal[3] = 1.0f / nug;
    }
    if (tid < KP) {
        float e2 = expf(theta[2]);
        sc[tid] = (tid < MNB) ? sqrtf(expf(-(float)(tid + 1) * e2)) : 0.0f;
    }
    __syncthreads();

    // ---- gather scaled neighbor responses: xs[rep][k] ----
    for (int idx = tid; idx < NREP * KP; idx += 256) {
        int rep = idx >> 5;
        int k   = idx & 31;
        float v = 0.0f;
        if (k < MNB) {
            int nb = cset[loc * MNB + k];
            if (nb >= 0) v = response[rep * NLOC + nb] * sc[k];
        }
        xs[rep * XS_S + k] = v;
    }
    if (tid < NREP) yv[tid] = response[tid * NLOC + loc];
    __syncthreads();

    if (tid < NREP) {
        float s = 0.0f;
        for (int k = 0; k < MNB; ++k) {
            float v = xs[tid * XS_S + k];
            s += v * v;
        }
        ss[tid] = s;
    }
    __syncthreads();

    // ---- lin = xs * xs^T via V_WMMA_F32_16X16X4_F32 ----
    // 8 waves, each computes 2 of the 16 (4x4) 16x16 output tiles.
    const int wave = tid >> 5;
    const int lane = tid & 31;
    const int half = lane >> 4;      // 0: lanes 0-15, 1: lanes 16-31
    const int lm   = lane & 15;
    const int I    = wave >> 1;              // row block 0..3
    const int J0   = (wave & 1) << 1;        // col blocks J0, J0+1
    const int J1   = J0 + 1;

    const int ra  = I  * 16 + lm;
    const int rb0 = J0 * 16 + lm;
    const int rb1 = J1 * 16 + lm;

    v8f c0 = {}; v8f c1 = {};
    for (int s = 0; s < 8; ++s) {
        int k0 = 4 * s + 2 * half;   // A/B layout: VGPR0 holds K=k0, VGPR1 holds K=k0+1
        v2f a, b0, b1;
        a.x  = xs[ra  * XS_S + k0];  a.y  = xs[ra  * XS_S + k0 + 1];
        b0.x = xs[rb0 * XS_S + k0];  b0.y = xs[rb0 * XS_S + k0 + 1];
        b1.x = xs[rb1 * XS_S + k0];  b1.y = xs[rb1 * XS_S + k0 + 1];
        c0 = __builtin_amdgcn_wmma_f32_16x16x4_f32(false, a, false, b0, (short)0, c0, false, false);
        c1 = __builtin_amdgcn_wmma_f32_16x16x4_f32(false, a, false, b1, (short)0, c1, false, false);
    }

    // ---- transform lin -> g = (lin + nonlin)/nug + I ----
    const float nug_inv  = scal[3];
    const float sig2     = scal[1];
    const float inv_len2 = scal[2];
    const float SQ3 = 1.7320508075688772f;
    for (int v = 0; v < 8; ++v) {
        int m = I * 16 + v + 8 * half;     // C/D layout: VGPR v -> row v (+8 for upper half)
        {
            int nn = J0 * 16 + lm;
            float lin = c0[v];
            float d2 = fmaxf((ss[m] + ss[nn] - 2.0f * lin) * inv_len2, 1e-12f);
            float d  = sqrtf(d2);
            float nl = sig2 * (1.0f + SQ3 * d) * expf(-SQ3 * d);
            g[m * G_S + nn] = (lin + nl) * nug_inv + ((m == nn) ? 1.0f : 0.0f);
        }
        {
            int nn = J1 * 16 + lm;
            float lin = c1[v];
            float d2 = fmaxf((ss[m] + ss[nn] - 2.0f * lin) * inv_len2, 1e-12f);
            float d  = sqrtf(d2);
            float nl = sig2 * (1.0f + SQ3 * d) * expf(-SQ3 * d);
            g[m * G_S + nn] = (lin + nl) * nug_inv + ((m == nn) ? 1.0f : 0.0f);
        }
    }
    __syncthreads();

    // ---- in-place Cholesky (lower) on g ----
    for (int k = 0; k < NREP; ++k) {
        if (tid == 0) g[k * G_S + k] = sqrtf(g[k * G_S + k]);
        __syncthreads();
        float dk = g[k * G_S + k];
        {
            int r = k + 1 + tid;
            if (r < NREP) g[r * G_S + k] /= dk;
        }
        __syncthreads();
        {
            int r = k + 1 + tid;
            if (r < NREP) {
                float lrk = g[r * G_S + k];
                for (int c = k + 1; c <= r; ++c)
                    g[r * G_S + c] -= lrk * g[c * G_S + k];
            }
        }
        __syncthreads();
    }

    // ---- forward substitution: L y~ = y ----
    for (int k = 0; k < NREP; ++k) {
        if (tid == 0) yv[k] /= g[k * G_S + k];
        __syncthreads();
        float yk = yv[k];
        int r = k + 1 + tid;
        if (r < NREP) yv[r] -= g[r * G_S + k] * yk;
        __syncthreads();
    }

    // ---- log-likelihood for this location ----
    if (tid == 0) {
        float ssum = 0.0f, logdet = 0.0f;
        for (int k = 0; k < NREP; ++k) {
            ssum   += yv[k] * yv[k];
            logdet += logf(g[k * G_S + k]);
        }
        float nug = scal[0];
        const float alpha      = 2.0625f;           // 1/NUG_MULT^2 + 2
        const float alpha_post = 34.0625f;          // alpha + n/2
        float beta      = nug * (alpha - 1.0f);
        float beta_post = beta + 0.5f * ssum;
        float ll = -logdet + alpha * logf(beta) - alpha_post * logf(beta_post)
                 + lgammaf(alpha_post) - lgammaf(alpha);
        partial[loc] = ll;
    }
}

// Deterministic single-block tree reduction -> scalar output
__global__ __launch_bounds__(256)
void vecchia_reduce_kernel(const float* __restrict__ partial, float* __restrict__ out)
{
    __shared__ float buf[256];
    float s = 0.0f;
    for (int i = threadIdx.x; i < NLOC; i += 256) s += partial[i];
    buf[threadIdx.x] = s;
    __syncthreads();
    for (int ofs = 128; ofs > 0; ofs >>= 1) {
        if (threadIdx.x < ofs) buf[threadIdx.x] += buf[threadIdx.x + ofs];
        __syncthreads();
    }
    if (threadIdx.x == 0) out[0] = -buf[0] / (float)NLOC;
}

extern "C" void kernel_launch(void* const* d_in, const int* in_sizes, int n_in,
                              void* d_out, int out_size, void* d_ws, size_t ws_size,
                              hipStream_t stream) {
    const float* locs     = (const float*)d_in[0];   // (20000, 2)
    const float* response = (const float*)d_in[1];   // (64, 20000)
    const int*   cset     = (const int*)  d_in[2];   // (20000, 30)
    const float* theta    = (const float*)d_in[3];   // (6,)
    float* partial = (float*)d_ws;                   // 20000 floats scratch
    (void)in_sizes; (void)n_in; (void)out_size; (void)ws_size;

    vecchia_loc_kernel<<<NLOC, 256, 0, stream>>>(locs, response, cset, theta, partial);
    vecchia_reduce_kernel<<<1, 256, 0, stream>>>(partial, (float*)d_out);
}